// QuanvolutionHybridQuantum_65481071402276
// MI455X (gfx1250) — compile-verified
//
#include <hip/hip_runtime.h>
#include <math.h>

typedef float v2f __attribute__((ext_vector_type(2)));
typedef float v8f __attribute__((ext_vector_type(8)));

// ---------------------------------------------------------------------------
// 4-qubit state-vector helpers (16 complex amps in registers).
// Flattened index: wire w <-> bit (3-w)  (wire 0 is most significant axis).
// ---------------------------------------------------------------------------
__device__ __forceinline__ void apply_rot(float* re, float* im, int wire, float theta, int kind) {
    const float h = 0.5f * theta;
    const float c = cosf(h), s = sinf(h);
    const int msk = 8 >> wire;
#pragma unroll
    for (int i = 0; i < 16; ++i) {
        if (i & msk) continue;
        const int j = i | msk;
        const float r0 = re[i], i0 = im[i], r1 = re[j], i1 = im[j];
        if (kind == 0) {            // RY
            re[i] = c * r0 - s * r1;  im[i] = c * i0 - s * i1;
            re[j] = s * r0 + c * r1;  im[j] = s * i0 + c * i1;
        } else if (kind == 1) {     // RX
            re[i] = c * r0 + s * i1;  im[i] = c * i0 - s * r1;
            re[j] = s * i0 + c * r1;  im[j] = -s * r0 + c * i1;
        } else {                    // RZ
            re[i] = c * r0 + s * i0;  im[i] = c * i0 - s * r0;
            re[j] = c * r1 - s * i1;  im[j] = c * i1 + s * r1;
        }
    }
}

__device__ __forceinline__ void apply_cnot(float* re, float* im, int ctrl, int tgt) {
    const int cm = 8 >> ctrl, tm = 8 >> tgt;
#pragma unroll
    for (int i = 0; i < 16; ++i) {
        if ((i & cm) && !(i & tm)) {
            const int j = i | tm;
            float tr = re[i], ti = im[i];
            re[i] = re[j]; im[i] = im[j];
            re[j] = tr;    im[j] = ti;
        }
    }
}

// Fixed RandomLayer: ry(0) rx(1) rz(2) cnot(0,1) ry(3) cnot(2,3) rx(0) cnot(1,2)
__device__ __forceinline__ void random_layer(float* re, float* im, const float* p) {
    apply_rot(re, im, 0, p[0], 0);
    apply_rot(re, im, 1, p[1], 1);
    apply_rot(re, im, 2, p[2], 2);
    apply_cnot(re, im, 0, 1);
    apply_rot(re, im, 3, p[3], 0);
    apply_cnot(re, im, 2, 3);
    apply_rot(re, im, 0, p[4], 1);
    apply_cnot(re, im, 1, 2);
}

// ---------------------------------------------------------------------------
// Setup: build Ur/Ui (16x16, row-major [n*16+k]) for filter & attn params,
// and the data-independent kernel scalar k = |(U_k^2)[0,0]|.
// Workspace layout (floats): [0]=UrF [256]=UiF [512]=UrA [768]=UiA [1024]=k
// ---------------------------------------------------------------------------
__global__ void quanv_setup(const float* __restrict__ fp, const float* __restrict__ ap,
                            const float* __restrict__ kp, float* __restrict__ ws) {
    const int t = threadIdx.x;
    if (t < 32) {
        const int col = t & 15;
        const float* prm = (t < 16) ? fp : ap;
        float* Ur = ws + ((t < 16) ? 0 : 512);
        float* Ui = Ur + 256;
        float re[16], im[16];
#pragma unroll
        for (int i = 0; i < 16; ++i) { re[i] = 0.f; im[i] = 0.f; }
        re[col] = 1.f;
        random_layer(re, im, prm);
#pragma unroll
        for (int n = 0; n < 16; ++n) { Ur[n * 16 + col] = re[n]; Ui[n * 16 + col] = im[n]; }
    } else if (t == 32) {
        float re[16], im[16];
#pragma unroll
        for (int i = 0; i < 16; ++i) { re[i] = 0.f; im[i] = 0.f; }
        re[0] = 1.f;
        random_layer(re, im, kp);
        random_layer(re, im, kp);
        ws[1024] = sqrtf(re[0] * re[0] + im[0] * im[0]);
    }
}

// ---------------------------------------------------------------------------
// One circuit evaluation for a 16-image tile via f32 WMMA.
// A-layout (16x4 f32): lane L(<16): M=L, VGPR0/1 = K=0/1 ; lanes 16-31: K=2/3.
// B-layout (4x16):     VGPR v: lanes 0-15 -> K=v, lanes 16-31 -> K=v+2, N=lane%16.
// B[k][n] = U[n][4kk+k]  ->  lane loads v2f at U + n*16 + 2*hi + 4*kk.
// Returns P = |U psi|^2 per element in D-layout: elem v of lane L holds
// p[image = v + 8*hi][amp = L%16].
// ---------------------------------------------------------------------------
__device__ __forceinline__ v8f quantum_probs(float a0, float a1, float a2, float a3,
                                             const float* __restrict__ Ur,
                                             const float* __restrict__ Ui,
                                             int m, int hi) {
    float cw[4], sw[4];
    cw[0] = cosf(0.5f * a0); sw[0] = sinf(0.5f * a0);
    cw[1] = cosf(0.5f * a1); sw[1] = sinf(0.5f * a1);
    cw[2] = cosf(0.5f * a2); sw[2] = sinf(0.5f * a2);
    cw[3] = cosf(0.5f * a3); sw[3] = sinf(0.5f * a3);
    // psi[i] = f(bit3:w0) f(bit2:w1) f(bit1:w2) f(bit0:w3); this lane needs
    // indices i = 4*kk + 2*hi + {0,1}: bit1 = hi, bits[3:2] = kk, bit0 = 0/1.
    const float f1 = hi ? sw[2] : cw[2];
    const int boff = m * 16 + 2 * hi;
    v8f Y = {}, Z = {};
#pragma unroll
    for (int kk = 0; kk < 4; ++kk) {
        const float f0 = (kk & 2) ? sw[0] : cw[0];
        const float f2 = (kk & 1) ? sw[1] : cw[1];
        const float base = f0 * f2 * f1;
        v2f A; A.x = base * cw[3]; A.y = base * sw[3];
        v2f Br = *(const v2f*)(Ur + boff + 4 * kk);
        v2f Bi = *(const v2f*)(Ui + boff + 4 * kk);
        Y = __builtin_amdgcn_wmma_f32_16x16x4_f32(false, A, false, Br, (short)0, Y, false, false);
        Z = __builtin_amdgcn_wmma_f32_16x16x4_f32(false, A, false, Bi, (short)0, Z, false, false);
    }
    v8f P;
#pragma unroll
    for (int e = 0; e < 8; ++e) P[e] = Y[e] * Y[e] + Z[e] * Z[e];
    return P;
}

// <Z_w> for image m from the wave's 16x16 probability tile in LDS.
__device__ __forceinline__ void reduce_feats(const float* __restrict__ tile, int m, float f[4]) {
    f[0] = f[1] = f[2] = f[3] = 0.f;
#pragma unroll
    for (int n = 0; n < 16; ++n) {
        const float pv = tile[m * 16 + n];
        f[0] += (n & 8) ? -pv : pv;
        f[1] += (n & 4) ? -pv : pv;
        f[2] += (n & 2) ? -pv : pv;
        f[3] += (n & 1) ? -pv : pv;
    }
}

// ---------------------------------------------------------------------------
// Fused main kernel: 1 wave = 16 images; filter circuit (patch (0,0) only;
// the other 195 patches are dead in the reference), attention circuit,
// linear head, log_softmax.
// ---------------------------------------------------------------------------
__global__ __launch_bounds__(256) void quanv_main(const float* __restrict__ x,
                                                  const float* __restrict__ W,
                                                  const float* __restrict__ bias,
                                                  const float* __restrict__ ws,
                                                  float* __restrict__ out, int B) {
    __shared__ float lds[8][256];              // per-wave 16x16 f32 tile
    const int lane = threadIdx.x & 31;
    const int wave = threadIdx.x >> 5;
    const int m    = lane & 15;                // image-in-tile this lane serves
    const int hi   = lane >> 4;                // A/B layout half selector
    const int base = (blockIdx.x * 8 + wave) * 16;
    const int imgRaw = base + m;
    const int img    = (imgRaw < B) ? imgRaw : (B - 1);
    float* tile = lds[wave];

    // Patch (0,0): pixels [0],[1],[28],[29] of the 28x28 image
    const float* px = x + (size_t)img * 784;
    const float a0 = px[0], a1 = px[1], a2 = px[28], a3 = px[29];

    // Circuit 1: quantum filter
    v8f P1 = quantum_probs(a0, a1, a2, a3, ws, ws + 256, m, hi);
#pragma unroll
    for (int v = 0; v < 8; ++v) tile[(v + 8 * hi) * 16 + m] = P1[v];
    __syncthreads();
    float feat[4];
    reduce_feats(tile, m, feat);
    __syncthreads();

    // Circuit 2: quantum self-attention on the 4 patch-0 features
    v8f P2 = quantum_probs(feat[0], feat[1], feat[2], feat[3], ws + 512, ws + 768, m, hi);
#pragma unroll
    for (int v = 0; v < 8; ++v) tile[(v + 8 * hi) * 16 + m] = P2[v];
    __syncthreads();

    if (hi == 0 && imgRaw < B) {
        float att[4];
        reduce_feats(tile, m, att);
        const float kval = ws[1024];
        float logits[10];
#pragma unroll
        for (int j = 0; j < 10; ++j) {
            logits[j] = bias[j]
                      + W[j * 5 + 0] * att[0] + W[j * 5 + 1] * att[1]
                      + W[j * 5 + 2] * att[2] + W[j * 5 + 3] * att[3]
                      + W[j * 5 + 4] * kval;
        }
        float mx = logits[0];
#pragma unroll
        for (int j = 1; j < 10; ++j) mx = fmaxf(mx, logits[j]);
        float sum = 0.f;
#pragma unroll
        for (int j = 0; j < 10; ++j) sum += expf(logits[j] - mx);
        const float lse = logf(sum) + mx;
        float* o = out + (size_t)imgRaw * 10;
#pragma unroll
        for (int j = 0; j < 10; ++j) o[j] = logits[j] - lse;
    }
}

extern "C" void kernel_launch(void* const* d_in, const int* in_sizes, int n_in,
                              void* d_out, int out_size, void* d_ws, size_t ws_size,
                              hipStream_t stream) {
    const float* x  = (const float*)d_in[0];   // (B, 28, 28)
    const float* fp = (const float*)d_in[1];   // filter_params (5)
    const float* ap = (const float*)d_in[2];   // attn_params   (5)
    const float* kp = (const float*)d_in[3];   // kernel_params (5)
    const float* W  = (const float*)d_in[4];   // (10, 5)
    const float* b  = (const float*)d_in[5];   // (10)
    float* ws  = (float*)d_ws;                 // needs 1025 floats
    float* out = (float*)d_out;                // (B, 10)

    const int B = in_sizes[0] / 784;

    quanv_setup<<<1, 64, 0, stream>>>(fp, ap, kp, ws);

    const int tiles  = (B + 15) / 16;          // 16 images per wave
    const int blocks = (tiles + 7) / 8;        // 8 waves per block
    quanv_main<<<blocks, 256, 0, stream>>>(x, W, b, ws, out, B);
}